// LSSTransform_71983651881170
// MI455X (gfx1250) — compile-verified
//
#include <hip/hip_runtime.h>

// ---------------- problem constants (match reference) ----------------
#define IN_H   450
#define IN_W   800
#define FHH    28
#define FWW    50
#define DCNT   59          // valid depth bins
#define DPAD   64          // padded to 4 WMMA M-tiles
#define BATCH  2
#define NCAM   6
#define CCH    80          // channels (5 WMMA N-tiles)
#define NXV    200
#define NYV    400
#define VOX    (NXV * NYV)                  // 80000
#define PIXB   (NCAM * FHH * FWW)           // 8400 pixels per batch
#define BLKB   (PIXB / 4)                   // 2100 blocks per batch (4 waves/blk)

typedef __attribute__((ext_vector_type(2))) float v2f;
typedef __attribute__((ext_vector_type(8))) float v8f;

// ---------------------------------------------------------------------
// Zero fill (ws or out accumulator). Grid-stride.
// ---------------------------------------------------------------------
__global__ void lss_zero_f32(float* __restrict__ p, size_t n) {
  size_t i = (size_t)blockIdx.x * blockDim.x + threadIdx.x;
  size_t stride = (size_t)gridDim.x * blockDim.x;
  for (; i < n; i += stride) p[i] = 0.0f;
}

// ---------------------------------------------------------------------
// Main lift+splat kernel. One wave32 per camera pixel; 4 waves / block.
// Batch index b derives from blockIdx.x ONLY -> the per-batch accumulator
// base stays in SGPRs and atomics lower to the GVS (saddr) form with a
// single 32-bit VGPR element offset (max offset 80000*80 < 2^31).
// Lift outer product depth[64] x feat[80] via V_WMMA_F32_16X16X4_F32
// (only K=0 populated => exact rank-1 outer product per 16x16 tile).
//   VOXMAJOR=1 : accum layout (B, VOX, C) -> coalesced atomics (fast path)
//   VOXMAJOR=0 : accum layout (B, C, VOX) -> direct into d_out (fallback)
// ---------------------------------------------------------------------
template <int VOXMAJOR>
__global__ __launch_bounds__(128) void lss_splat(
    const float* __restrict__ img_feats,    // (B,N,C,FH,FW)
    const float* __restrict__ depth_probs,  // (B,N,D,FH,FW)
    const float* __restrict__ sensor2ego,   // (B,N,4,4)
    const float* __restrict__ intrinsics,   // (B,N,3,3)
    float* __restrict__ accum)
{
  __shared__ float sDepth[4][DPAD];
  __shared__ float sFeat[4][CCH];
  __shared__ int   sVox[4][DPAD];

  const int lane = threadIdx.x & 31;
  const int wv   = threadIdx.x >> 5;

  // b from blockIdx only => wave-uniform, provably scalar.
  const int b    = blockIdx.x / BLKB;
  const int pixL = (blockIdx.x % BLKB) * 4 + wv;   // pixel within batch

  const int w = pixL % FWW;
  const int h = (pixL / FWW) % FHH;
  const int n = pixL / (FWW * FHH);
  const int cam = b * NCAM + n;

  // ---- per-camera matrices ----
  const float* Ki = intrinsics + (size_t)cam * 9;
  const float m00 = Ki[0], m01 = Ki[1], m02 = Ki[2];
  const float m10 = Ki[3], m11 = Ki[4], m12 = Ki[5];
  const float m20 = Ki[6], m21 = Ki[7], m22 = Ki[8];
  const float det = m00 * (m11 * m22 - m12 * m21)
                  - m01 * (m10 * m22 - m12 * m20)
                  + m02 * (m10 * m21 - m11 * m20);
  const float rdet = 1.0f / det;
  const float i00 = (m11 * m22 - m12 * m21) * rdet;
  const float i01 = (m02 * m21 - m01 * m22) * rdet;
  const float i02 = (m01 * m12 - m02 * m11) * rdet;
  const float i10 = (m12 * m20 - m10 * m22) * rdet;
  const float i11 = (m00 * m22 - m02 * m20) * rdet;
  const float i12 = (m02 * m10 - m00 * m12) * rdet;
  const float i20 = (m10 * m21 - m11 * m20) * rdet;
  const float i21 = (m01 * m20 - m00 * m21) * rdet;
  const float i22 = (m00 * m11 - m01 * m10) * rdet;

  const float* S = sensor2ego + (size_t)cam * 16;
  const float s00 = S[0], s01 = S[1],  s02 = S[2],  s03 = S[3];
  const float s10 = S[4], s11 = S[5],  s12 = S[6],  s13 = S[7];
  const float s20 = S[8], s21 = S[9],  s22 = S[10], s23 = S[11];

  // ---- frustum pixel coords (linspace over full image) ----
  const float x = (float)w * ((float)(IN_W - 1) / (float)(FWW - 1));
  const float y = (float)h * ((float)(IN_H - 1) / (float)(FHH - 1));

  // camera-frame ray direction (Kinv @ [x,y,1]); depth scales all 3 comps
  const float rx = i00 * x + i01 * y + i02;
  const float ry = i10 * x + i11 * y + i12;
  const float rz = i20 * x + i21 * y + i22;

  const size_t pixOff  = (size_t)h * FWW + w;
  const float* dp_base = depth_probs + ((size_t)cam * DCNT) * (FHH * FWW) + pixOff;
  const float* ft_base = img_feats   + ((size_t)cam * CCH ) * (FHH * FWW) + pixOff;

  // ---- geometry + depth staging: lane handles d = lane, lane+32 ----
  #pragma unroll
  for (int dd = 0; dd < 2; ++dd) {
    const int d = lane + 32 * dd;
    const float depth = (float)(d + 1);
    const float px = rx * depth, py = ry * depth, pz = rz * depth;
    const float ex = s00 * px + s01 * py + s02 * pz + s03;
    const float ey = s10 * px + s11 * py + s12 * pz + s13;
    const float ez = s20 * px + s21 * py + s22 * pz + s23;
    const bool kept = (ex >= -15.0f) & (ex < 15.0f) &
                      (ey >= -30.0f) & (ey < 30.0f) &
                      (ez >=  -2.0f) & (ez <  2.0f) & (d < DCNT);
    int ix = (int)floorf((ex + 15.0f) * (1.0f / 0.15f));
    int iy = (int)floorf((ey + 30.0f) * (1.0f / 0.15f));
    ix = ix < 0 ? 0 : (ix > NXV - 1 ? NXV - 1 : ix);
    iy = iy < 0 ? 0 : (iy > NYV - 1 ? NYV - 1 : iy);
    // pre-scale voxel index into an element offset for the scatter
    const int voff = VOXMAJOR ? (iy * NXV + ix) * CCH : (iy * NXV + ix);
    sVox[wv][d]   = kept ? voff : -1;
    sDepth[wv][d] = (d < DCNT) ? dp_base[(size_t)d * (FHH * FWW)] : 0.0f;
  }

  // ---- feature staging ----
  for (int c = lane; c < CCH; c += 32)
    sFeat[wv][c] = ft_base[(size_t)c * (FHH * FWW)];

  __syncthreads();

  // ---- lift (WMMA outer product) + splat (coalesced f32 atomics) ----
  const int  cLane   = lane & 15;
  const bool lowHalf = lane < 16;
  float* const accB  = accum + (size_t)b * (VOX * CCH);  // SGPR base

  for (int mt = 0; mt < 4; ++mt) {
    // Hoist this lane's 8 voxel offsets into registers (2x ds_load_b128),
    // reused across all 5 channel blocks.
    const int mbase = mt * 16 + (lowHalf ? 0 : 8);
    int voxr[8];
    #pragma unroll
    for (int v = 0; v < 8; ++v) voxr[v] = sVox[wv][mbase + v];

    v2f a;
    a.x = lowHalf ? sDepth[wv][mt * 16 + cLane] : 0.0f;  // A[M][K=0]
    a.y = 0.0f;                                          // all other K zero

    for (int cb = 0; cb < 5; ++cb) {
      v2f bm;
      bm.x = lowHalf ? sFeat[wv][cb * 16 + cLane] : 0.0f; // B[K=0][N]
      bm.y = 0.0f;
      v8f acc = {};
      // D[M][N] = depth[mt*16+M] * feat[cb*16+N]  (exact rank-1 product)
      acc = __builtin_amdgcn_wmma_f32_16x16x4_f32(
          /*neg_a=*/false, a, /*neg_b=*/false, bm,
          /*c_mod=*/(short)0, acc, /*reuse_a=*/false, /*reuse_b=*/false);

      const int c = cb * 16 + cLane;
      const int cOff = VOXMAJOR ? c : c * VOX;
      #pragma unroll
      for (int v = 0; v < 8; ++v) {
        if (voxr[v] >= 0) {
          // u32 element offset from uniform SGPR base -> saddr-form atomic
          atomicAdd(accB + (unsigned)(voxr[v] + cOff), acc[v]);
        }
      }
    }
  }
}

// ---------------------------------------------------------------------
// (B, VOX, C) scratch  ->  (B, C, VOX) output. LDS-tiled transpose,
// coalesced on both global sides, 81-stride padding kills bank conflicts.
// ---------------------------------------------------------------------
__global__ __launch_bounds__(256) void lss_transpose(
    const float* __restrict__ ws, float* __restrict__ out)
{
  __shared__ float tile[32 * 81];
  const int b    = blockIdx.y;
  const int vox0 = blockIdx.x * 32;
  const float* src = ws + ((size_t)b * VOX + vox0) * CCH;
  for (int i = threadIdx.x; i < 32 * CCH; i += 256) {
    const int vs = i / CCH, c = i % CCH;
    tile[vs * 81 + c] = src[i];
  }
  __syncthreads();
  float* dst = out + (size_t)b * CCH * VOX + vox0;
  for (int i = threadIdx.x; i < 32 * CCH; i += 256) {
    const int c = i >> 5, vs = i & 31;
    dst[(size_t)c * VOX + vs] = tile[vs * 81 + c];
  }
}

// ---------------------------------------------------------------------
extern "C" void kernel_launch(void* const* d_in, const int* in_sizes, int n_in,
                              void* d_out, int out_size, void* d_ws, size_t ws_size,
                              hipStream_t stream) {
  const float* img_feats   = (const float*)d_in[0];
  const float* depth_probs = (const float*)d_in[1];
  const float* sensor2ego  = (const float*)d_in[2];
  const float* intrinsics  = (const float*)d_in[3];
  float* out = (float*)d_out;

  const size_t nelem = (size_t)BATCH * VOX * CCH;     // 12.8M floats
  const size_t need  = nelem * sizeof(float);

  if (ws_size >= need) {
    // Fast path: voxel-major accumulation in scratch, then transpose.
    float* ws = (float*)d_ws;
    lss_zero_f32<<<2048, 256, 0, stream>>>(ws, nelem);
    lss_splat<1><<<BATCH * BLKB, 128, 0, stream>>>(img_feats, depth_probs,
                                                   sensor2ego, intrinsics, ws);
    dim3 g(VOX / 32, BATCH);
    lss_transpose<<<g, 256, 0, stream>>>(ws, out);
  } else {
    // Fallback: accumulate directly into d_out (channel-major atomics).
    lss_zero_f32<<<2048, 256, 0, stream>>>(out, nelem);
    lss_splat<0><<<BATCH * BLKB, 128, 0, stream>>>(img_feats, depth_probs,
                                                   sensor2ego, intrinsics, out);
  }
}